// SelectiveLinear_63711544869389
// MI455X (gfx1250) — compile-verified
//
#include <hip/hip_runtime.h>

// Problem constants (match reference)
#define B_    8
#define S_    2048
#define NEXP_ 16
#define DIN_  1024
#define DOUT_ 1024

// GEMM tiling
#define KC_     64                 // K-chunk staged in LDS per double-buffer slot
#define LDSROWS 64                 // DOUT rows (i-tile) staged per block
#define LDSPAD  4                  // pad to 68 floats/row -> conflict-free ds_load_b64
#define LDSSTR  (KC_ + LDSPAD)     // 68

typedef __attribute__((ext_vector_type(2))) float v2f;
typedef __attribute__((ext_vector_type(8))) float v8f;

// s_wait_asynccnt <= N (ASYNCcnt tracks GLOBAL_LOAD_ASYNC_TO_LDS_*)
#define WAIT_ASYNCCNT(N) asm volatile("s_wait_asynccnt %0" :: "n"(N) : "memory")

// ---------------------------------------------------------------------------
// Stage 1a: mixed_w[b,i,j] = sum_n p[b,n] * W[n,i,j]
// One thread per (i,j); reads the 16 expert values once, emits all 8 batches.
// ---------------------------------------------------------------------------
__global__ __launch_bounds__(256)
void mix_weights_kernel(const float* __restrict__ p,    // [B, NEXP]
                        const float* __restrict__ w,    // [NEXP, DOUT, DIN]
                        float* __restrict__ mw)         // [B, DOUT, DIN]
{
    const int idx = blockIdx.x * 256 + threadIdx.x;     // over DOUT*DIN
    float wv[NEXP_];
#pragma unroll
    for (int n = 0; n < NEXP_; ++n)
        wv[n] = w[(size_t)n * (DOUT_ * DIN_) + idx];

#pragma unroll
    for (int b = 0; b < B_; ++b) {
        float acc = 0.0f;
#pragma unroll
        for (int n = 0; n < NEXP_; ++n)
            acc += p[b * NEXP_ + n] * wv[n];            // p is uniform -> s_loads
        mw[(size_t)b * (DOUT_ * DIN_) + idx] = acc;
    }
}

// ---------------------------------------------------------------------------
// Stage 1b: mixed_b[b,i] = sum_n p[b,n] * bias[n,i]
// ---------------------------------------------------------------------------
__global__ __launch_bounds__(256)
void mix_bias_kernel(const float* __restrict__ p,       // [B, NEXP]
                     const float* __restrict__ bias,    // [NEXP, DOUT]
                     float* __restrict__ mb)            // [B, DOUT]
{
    const int i = blockIdx.x * 256 + threadIdx.x;       // over DOUT
    float bv[NEXP_];
#pragma unroll
    for (int n = 0; n < NEXP_; ++n)
        bv[n] = bias[n * DOUT_ + i];
#pragma unroll
    for (int b = 0; b < B_; ++b) {
        float acc = 0.0f;
#pragma unroll
        for (int n = 0; n < NEXP_; ++n)
            acc += p[b * NEXP_ + n] * bv[n];
        mb[b * DOUT_ + i] = acc;
    }
}

// ---------------------------------------------------------------------------
// Stage 2: out[b,s,i] = sum_j mw[b,i,j] * x[b,s,j] + mb[b,i]
//
// Block: 256 threads = 8 waves. Block tile = 256 (S) x 64 (DOUT).
// Each wave: 32 (S) x 64 (DOUT) via fp32 WMMA 16x16x4, 8 accumulators.
// mw i-tile (64 rows x KC cols) is staged in LDS by async global->LDS copies,
// double-buffered, shared by all 8 waves. x fragments stream from global/L2.
//
// Fragment layouts (ISA 7.12.2):
//   A (x):  lane -> row M = lane&15, cols K = 2*(lane>>4)+{0,1}  (float2 load)
//   B (mw): lane -> col N = lane&15, rows K = 2*(lane>>4)+{0,1}  (ds b64 load)
//   D: VGPR v of lane L holds (M = v + 8*(L>>4), N = L&15)
// ---------------------------------------------------------------------------
__global__ __launch_bounds__(256)
void selgemm_kernel(const float* __restrict__ x,    // [B, S, DIN]
                    const float* __restrict__ mw,   // [B, DOUT, DIN]
                    const float* __restrict__ mb,   // [B, DOUT]
                    float* __restrict__ out)        // [B, S, DOUT]
{
    __shared__ float smem[2][LDSROWS * LDSSTR];

    const int tid  = threadIdx.x;
    const int lane = tid & 31;
    const int wave = tid >> 5;

    const int b  = blockIdx.z;
    const int i0 = blockIdx.x * 64;                 // DOUT tile origin
    const int s0 = blockIdx.y * 256 + wave * 32;    // S origin for this wave

    const int m    = lane & 15;                     // A row / B,D column index
    const int half = lane >> 4;
    const int kb   = half * 2;                      // K sub-offset in 4-wide step

    // ---- async-copy setup: block cooperatively stages 64 x KC mw tile ----
    const int r  = tid >> 2;                        // 0..63: tile row
    const int c0 = (tid & 3) << 4;                  // 0/16/32/48: first col
    const float*   sbase = mw + (size_t)(b * DOUT_ + i0) * DIN_;      // uniform
    const unsigned vgoff = (unsigned)((r * DIN_ + c0) * sizeof(float));
    const unsigned lbase0 = (unsigned)(size_t)&smem[0][r * LDSSTR + c0];
    const unsigned lbase1 = (unsigned)(size_t)&smem[1][r * LDSSTR + c0];

    auto issue_chunk = [&](int kc) {
        const unsigned go = vgoff + (unsigned)(kc * KC_ * sizeof(float));
        const unsigned la = (kc & 1) ? lbase1 : lbase0;
        // 4 x b128 per thread = 64 B; block total = 16 KB tile. ASYNCcnt += 4.
        asm volatile(
            "global_load_async_to_lds_b128 %0, %1, %2 offset:0\n\t"
            "global_load_async_to_lds_b128 %0, %1, %2 offset:16\n\t"
            "global_load_async_to_lds_b128 %0, %1, %2 offset:32\n\t"
            "global_load_async_to_lds_b128 %0, %1, %2 offset:48"
            :: "v"(la), "v"(go), "s"(sbase) : "memory");
    };

    // ---- A-operand row pointers (8-byte aligned) ----
    const float* xrow0 = x + ((size_t)(b * S_ + s0 + m) * DIN_) + kb;
    const float* xrow1 = xrow0 + (size_t)16 * DIN_;

    v8f acc[2][4];
#pragma unroll
    for (int f = 0; f < 2; ++f)
#pragma unroll
        for (int t = 0; t < 4; ++t)
            acc[f][t] = (v8f){};

    issue_chunk(0);

    const int NC = DIN_ / KC_;                      // 16 chunks
    for (int kc = 0; kc < NC; ++kc) {
        if (kc + 1 < NC) {
            issue_chunk(kc + 1);                    // prefetch next buffer
            WAIT_ASYNCCNT(4);                       // current chunk landed
        } else {
            WAIT_ASYNCCNT(0);
        }
        __syncthreads();                            // all waves' copies visible

        const float* bb  = &smem[kc & 1][m * LDSSTR + kb];  // + t*16*LDSSTR
        const float* xa0 = xrow0 + kc * KC_;
        const float* xa1 = xrow1 + kc * KC_;

#pragma unroll 4
        for (int k = 0; k < KC_; k += 4) {
            const v2f a0 = *(const v2f*)(xa0 + k);
            const v2f a1 = *(const v2f*)(xa1 + k);
#pragma unroll
            for (int t = 0; t < 4; ++t) {
                const v2f bt = *(const v2f*)(bb + t * (16 * LDSSTR) + k);
                acc[0][t] = __builtin_amdgcn_wmma_f32_16x16x4_f32(
                                false, a0, false, bt, (short)0, acc[0][t], false, false);
                acc[1][t] = __builtin_amdgcn_wmma_f32_16x16x4_f32(
                                false, a1, false, bt, (short)0, acc[1][t], false, false);
            }
        }
        __syncthreads();                            // done reading this buffer
    }

    // ---- epilogue: bias add + store; lane owns column (i0 + t*16 + m) ----
    float biasv[4];
#pragma unroll
    for (int t = 0; t < 4; ++t)
        biasv[t] = mb[b * DOUT_ + i0 + t * 16 + m];

#pragma unroll
    for (int f = 0; f < 2; ++f) {
#pragma unroll
        for (int v = 0; v < 8; ++v) {
            const size_t row = (size_t)(b * S_ + s0 + f * 16 + v + 8 * half) * DOUT_;
#pragma unroll
            for (int t = 0; t < 4; ++t)
                out[row + i0 + t * 16 + m] = acc[f][t][v] + biasv[t];
        }
    }
}

// ---------------------------------------------------------------------------
// Host-side launcher
// ---------------------------------------------------------------------------
extern "C" void kernel_launch(void* const* d_in, const int* in_sizes, int n_in,
                              void* d_out, int out_size, void* d_ws, size_t ws_size,
                              hipStream_t stream) {
    (void)in_sizes; (void)n_in; (void)out_size; (void)ws_size;

    const float* x    = (const float*)d_in[0];   // [B, S, DIN]
    const float* p    = (const float*)d_in[1];   // [B, NEXP]
    const float* w    = (const float*)d_in[2];   // [NEXP, DOUT, DIN]
    const float* bias = (const float*)d_in[3];   // [NEXP, DOUT]
    float*       out  = (float*)d_out;           // [B, S, DOUT]

    // Workspace: mixed_w [B,DOUT,DIN] fp32 (32 MB), then mixed_b [B,DOUT].
    float* mw = (float*)d_ws;
    float* mb = (float*)((char*)d_ws + (size_t)B_ * DOUT_ * DIN_ * sizeof(float));

    mix_weights_kernel<<<dim3((DOUT_ * DIN_) / 256), dim3(256), 0, stream>>>(p, w, mw);
    mix_bias_kernel<<<dim3(DOUT_ / 256), dim3(256), 0, stream>>>(p, bias, mb);

    dim3 grid(DOUT_ / 64, S_ / 256, B_);         // (16, 8, 8)
    selgemm_kernel<<<grid, dim3(256), 0, stream>>>(x, mw, mb, out);
}